// LinearAttention_23948737643275
// MI455X (gfx1250) — compile-verified
//
#include <hip/hip_runtime.h>

// ---------------------------------------------------------------------------
// Linear attention (fp32) for MI455X / gfx1250, built on V_WMMA_F32_16X16X4_F32.
//   q = relu(Q @ Wq), k = relu(K @ Wk)          (per-head 64x64 projections)
//   S = tril(q k^T);  attn = S / rowsum(S);  out = attn @ V
// Outputs: d_out = [ out (B*H*L*D) | attn (B*H*L*L) ]  (fp32)
// Workspace: Qf (16 MB) + KfT (16 MB) = 32 MB in d_ws.
// ---------------------------------------------------------------------------

typedef __attribute__((ext_vector_type(2))) float v2f;
typedef __attribute__((ext_vector_type(8))) float v8f;

#define B_    4
#define H_    16
#define L_    1024
#define D_    64
#define ROWS  32            // rows of S per block (S strip 32 x 1024 fits LDS)
#define NRB   (L_ / ROWS)   // 32 row-blocks per (b,h)
#define CB    64            // column-block width

// One fp32 WMMA step: D(16x16) += A(16x4) * B(4x16).
// A per-lane: M = lane&15, K = {0,1} (lanes 0-15) / {2,3} (lanes 16-31)
// B per-lane: N = lane&15, rows K same half-wave split
// D per-lane: N = lane&15, M = vgpr + (lane>=16 ? 8 : 0)
__device__ __forceinline__ v8f wmma_f32_step(float a0, float a1,
                                             float b0, float b1, v8f c) {
  v2f a; a[0] = a0; a[1] = a1;
  v2f b; b[0] = b0; b[1] = b1;
  return __builtin_amdgcn_wmma_f32_16x16x4_f32(
      /*neg_a=*/false, a, /*neg_b=*/false, b,
      /*c_mod=*/(short)0, c, /*reuse_a=*/false, /*reuse_b=*/false);
}

// ---------------------------------------------------------------------------
// Kernel 1: per-head feature map  Y = relu(X @ W), W is [H][64][64].
// TR==0: Y stored row-major   Yf [bh][l][d]   (Qf, used as WMMA A later)
// TR==1: Y stored transposed  YfT[bh][d][l]   (KfT, used as WMMA B later)
// Block: 256 threads = 8 waves; each wave owns one 16x16 tile of the 32x64 out.
// ---------------------------------------------------------------------------
template <int TR>
__global__ __launch_bounds__(256) void fm_kernel(const float* __restrict__ X,
                                                 const float* __restrict__ W,
                                                 float* __restrict__ Y) {
  __shared__ float Wbuf[64][72];  // stride 72 floats -> conflict-free B reads

  const int rb = blockIdx.x & (NRB - 1);
  const int bh = blockIdx.x >> 5;          // NRB == 32
  const int h  = bh & (H_ - 1);
  const int tid = threadIdx.x;

  const float* Wh = W + (size_t)h * D_ * D_;
  for (int i = tid; i < D_ * D_; i += 256)
    Wbuf[i >> 6][i & 63] = Wh[i];
  __syncthreads();

  const int lane = tid & 31, wid = tid >> 5;
  const int half = lane >> 4, ln = lane & 15, koff = half * 2;
  const int mt = wid >> 2, nt = wid & 3;   // 2 x 4 tile grid over 32 x 64

  const float* Xrow = X + ((size_t)bh * L_ + rb * ROWS + mt * 16 + ln) * D_;

  v8f acc = {};
#pragma unroll
  for (int k = 0; k < D_; k += 4) {
    float a0 = Xrow[k + koff];
    float a1 = Xrow[k + koff + 1];
    float b0 = Wbuf[k + koff][nt * 16 + ln];
    float b1 = Wbuf[k + koff + 1][nt * 16 + ln];
    acc = wmma_f32_step(a0, a1, b0, b1, acc);
  }

  const int e = nt * 16 + ln;
#pragma unroll
  for (int v = 0; v < 8; ++v) {
    float val = fmaxf(acc[v], 0.0f);       // ReLU
    int r  = mt * 16 + half * 8 + v;       // local row
    int gl = rb * ROWS + r;                // global l
    if (TR)
      Y[((size_t)bh * D_ + e) * L_ + gl] = val;   // KfT[bh][d][l]
    else
      Y[((size_t)bh * L_ + gl) * D_ + e] = val;   // Qf [bh][l][d]
  }
}

// ---------------------------------------------------------------------------
// Kernel 2: per (b,h, 32-row strip):
//   S(32 x ncols) = Qf * KfT  (causal-masked)  -> kept entirely in LDS
//   rowsum -> inv; write normalized attn (with zero upper triangle)
//   O = (S @ V) * inv  -> out
// 256 threads = 8 waves; wave (mt,nt) owns 16x16 tiles.
// ---------------------------------------------------------------------------
__global__ __launch_bounds__(256) void attn_kernel(const float* __restrict__ Qf,
                                                   const float* __restrict__ KfT,
                                                   const float* __restrict__ V,
                                                   float* __restrict__ Oout,
                                                   float* __restrict__ Aout) {
  __shared__ float Sbuf[ROWS][1028];   // 128.5 KB, stride 1028 (bank shift 4/row)
  __shared__ float Qbuf[ROWS][68];     //   8.5 KB, stride 68
  __shared__ float KVbuf[64][72];      //  18.0 KB, stride 72 (Kf tile, then V tile)
  __shared__ float inv_[ROWS];

  const int rb = blockIdx.x & (NRB - 1);
  const int bh = blockIdx.x >> 5;
  const int tid = threadIdx.x;
  const int lane = tid & 31, wid = tid >> 5;
  const int half = lane >> 4, ln = lane & 15, koff = half * 2;
  const int mt = wid >> 2, nt = wid & 3;   // 2 x 4 tiles over 32 x 64

  // Load Qf strip (32 x 64) once.
  {
    const float* src = Qf + ((size_t)bh * L_ + rb * ROWS) * D_;
    for (int i = tid; i < ROWS * D_; i += 256)
      Qbuf[i >> 6][i & 63] = src[i];
  }

  const int grmax = rb * ROWS + ROWS - 1;
  const int nct   = grmax / CB + 1;        // causal column blocks
  const int ncols = nct * CB;

  // ---- Phase 1: S strip into LDS (masked, unnormalized) ----
  for (int ct = 0; ct < nct; ++ct) {
    __syncthreads();
    for (int i = tid; i < CB * D_; i += 256)   // KfT tile: [d][l] 64x64
      KVbuf[i >> 6][i & 63] =
          KfT[((size_t)bh * D_ + (i >> 6)) * L_ + ct * CB + (i & 63)];
    __syncthreads();

    v8f acc = {};
#pragma unroll
    for (int k = 0; k < D_; k += 4) {
      float a0 = Qbuf[mt * 16 + ln][k + koff];
      float a1 = Qbuf[mt * 16 + ln][k + koff + 1];
      float b0 = KVbuf[k + koff][nt * 16 + ln];
      float b1 = KVbuf[k + koff + 1][nt * 16 + ln];
      acc = wmma_f32_step(a0, a1, b0, b1, acc);
    }

    const int cl = nt * 16 + ln;
    const int gc = ct * CB + cl;
#pragma unroll
    for (int v = 0; v < 8; ++v) {
      int r  = mt * 16 + half * 8 + v;
      int gr = rb * ROWS + r;
      Sbuf[r][ct * CB + cl] = (gc > gr) ? 0.0f : acc[v];
    }
  }
  __syncthreads();

  // ---- Phase 2: row sums -> 1/sum ----
  {
    int r = tid >> 3, p = tid & 7;           // 8 threads per row
    float s = 0.0f;
    for (int c = p; c < ncols; c += 8) s += Sbuf[r][c];
    s += __shfl_xor(s, 1);
    s += __shfl_xor(s, 2);
    s += __shfl_xor(s, 4);
    if (p == 0) inv_[r] = 1.0f / s;
  }
  __syncthreads();

  // ---- Phase 3: write normalized attn (zero-fill beyond causal blocks) ----
  {
    const size_t abase = ((size_t)bh * L_ + rb * ROWS) * L_;
    for (int i = tid; i < ROWS * L_; i += 256) {
      int r = i >> 10, c = i & (L_ - 1);
      float val = (c < ncols) ? Sbuf[r][c] * inv_[r] : 0.0f;
      Aout[abase + (size_t)r * L_ + c] = val;
    }
  }

  // ---- Phase 4: O = (S @ V) * inv ----
  v8f oacc = {};
  for (int ct = 0; ct < nct; ++ct) {
    __syncthreads();
    for (int i = tid; i < CB * D_; i += 256)   // V tile: [l][d] 64x64
      KVbuf[i >> 6][i & 63] =
          V[((size_t)bh * L_ + ct * CB + (i >> 6)) * D_ + (i & 63)];
    __syncthreads();

#pragma unroll
    for (int k = 0; k < CB; k += 4) {
      float a0 = Sbuf[mt * 16 + ln][ct * CB + k + koff];
      float a1 = Sbuf[mt * 16 + ln][ct * CB + k + koff + 1];
      float b0 = KVbuf[k + koff][nt * 16 + ln];
      float b1 = KVbuf[k + koff + 1][nt * 16 + ln];
      oacc = wmma_f32_step(a0, a1, b0, b1, oacc);
    }
  }

#pragma unroll
  for (int v = 0; v < 8; ++v) {
    int r = mt * 16 + half * 8 + v;
    Oout[((size_t)bh * L_ + rb * ROWS + r) * D_ + nt * 16 + ln] =
        oacc[v] * inv_[r];
  }
}

// ---------------------------------------------------------------------------
extern "C" void kernel_launch(void* const* d_in, const int* in_sizes, int n_in,
                              void* d_out, int out_size, void* d_ws, size_t ws_size,
                              hipStream_t stream) {
  const float* q  = (const float*)d_in[0];   // [B,H,L,D]
  const float* k  = (const float*)d_in[1];   // [B,H,L,D]
  const float* v  = (const float*)d_in[2];   // [B,H,L,D]
  const float* wq = (const float*)d_in[3];   // [H,D,D]
  const float* wk = (const float*)d_in[4];   // [H,D,D]

  float* out  = (float*)d_out;                              // B*H*L*D
  float* attn = out + (size_t)B_ * H_ * L_ * D_;            // B*H*L*L

  float* Qf  = (float*)d_ws;                                // 16 MB
  float* KfT = Qf + (size_t)B_ * H_ * L_ * D_;              // 16 MB

  const dim3 blk(256);
  const dim3 grd(B_ * H_ * NRB);   // 2048 blocks

  fm_kernel<0><<<grd, blk, 0, stream>>>(q, wq, Qf);
  fm_kernel<1><<<grd, blk, 0, stream>>>(k, wk, KfT);
  attn_kernel<<<grd, blk, 0, stream>>>(Qf, KfT, v, out, attn);

  (void)in_sizes; (void)n_in; (void)out_size; (void)ws_size;
}